// TrainDataAugmentation_11020886082106
// MI455X (gfx1250) — compile-verified
//
#include <hip/hip_runtime.h>
#include <hip/hip_bf16.h>

typedef __attribute__((ext_vector_type(2))) float v2f;
typedef __attribute__((ext_vector_type(8))) float v8f;
// dword-aligned 2-float vector for b64 loads at 4-byte-aligned addresses
typedef __attribute__((ext_vector_type(2), aligned(4))) float v2f_u;

// ---------------------------------------------------------------------------
// Setup kernel: per-group (n=28) math. Builds R1/R2 (euler xyz), translation
// rows u1/u2 for M1/M2, writes the 12 useful floats of each matrix to ws,
// and computes dof6 = se3_log(inv(M1) @ M2) directly into d_out tail.
// Runs once as a single tiny block; its cost is irrelevant.
// ---------------------------------------------------------------------------
__device__ __forceinline__ void mat3_mul(const float* A, const float* B, float* Cm) {
#pragma unroll
    for (int i = 0; i < 3; ++i)
#pragma unroll
        for (int j = 0; j < 3; ++j)
            Cm[i * 3 + j] = A[i * 3 + 0] * B[0 * 3 + j] +
                            A[i * 3 + 1] * B[1 * 3 + j] +
                            A[i * 3 + 2] * B[2 * 3 + j];
}

__device__ __forceinline__ void euler_xyz(float ax, float ay, float az, float* R) {
    float cx = cosf(ax), sx = sinf(ax);
    float cy = cosf(ay), sy = sinf(ay);
    float cz = cosf(az), sz = sinf(az);
    float Rx[9] = {1.f, 0.f, 0.f,   0.f, cx, -sx,   0.f, sx, cx};
    float Ry[9] = {cy, 0.f, sy,     0.f, 1.f, 0.f,  -sy, 0.f, cy};
    float Rz[9] = {cz, -sz, 0.f,    sz, cz, 0.f,    0.f, 0.f, 1.f};
    float T[9];
    mat3_mul(Rx, Ry, T);
    mat3_mul(T, Rz, R);
}

__global__ void tda_setup_kernel(const float* __restrict__ C,
                                 const float* __restrict__ ang1,
                                 const float* __restrict__ ang2,
                                 const float* __restrict__ t1,
                                 const float* __restrict__ t2,
                                 float* __restrict__ mats,   // (2*n)*12 floats
                                 float* __restrict__ dof6,   // n*6 floats
                                 int n) {
    int g = blockIdx.x * blockDim.x + threadIdx.x;
    if (g >= n) return;

    float Cv[3] = {C[g * 3 + 0], C[g * 3 + 1], C[g * 3 + 2]};
    float R1[9], R2[9];
    euler_xyz(ang1[g * 3 + 0], ang1[g * 3 + 1], ang1[g * 3 + 2], R1);
    euler_xyz(ang2[g * 3 + 0], ang2[g * 3 + 1], ang2[g * 3 + 2], R2);

    // M = T(-C) @ Rot @ T(t) @ T(C):  linear = R, trans row u = -C@R + t + C
    float u1[3], u2[3];
#pragma unroll
    for (int j = 0; j < 3; ++j) {
        u1[j] = -(Cv[0] * R1[0 * 3 + j] + Cv[1] * R1[1 * 3 + j] + Cv[2] * R1[2 * 3 + j])
                + t1[g * 3 + j] + Cv[j];
        u2[j] = -(Cv[0] * R2[0 * 3 + j] + Cv[1] * R2[1 * 3 + j] + Cv[2] * R2[2 * 3 + j])
                + t2[g * 3 + j] + Cv[j];
    }

    // Stash M1, M2 (rows k=0..3 x cols j=0..2) for the streaming kernel.
    float* m1 = mats + (long long)g * 12;
    float* m2 = mats + (long long)(n + g) * 12;
#pragma unroll
    for (int k = 0; k < 3; ++k)
#pragma unroll
        for (int j = 0; j < 3; ++j) {
            m1[k * 3 + j] = R1[k * 3 + j];
            m2[k * 3 + j] = R2[k * 3 + j];
        }
#pragma unroll
    for (int j = 0; j < 3; ++j) { m1[9 + j] = u1[j]; m2[9 + j] = u2[j]; }

    // F = inv(M1) @ M2 (rigid inverse: linear R1^T, trans -u1 @ R1^T)
    float Rf[9];
#pragma unroll
    for (int i = 0; i < 3; ++i)
#pragma unroll
        for (int j = 0; j < 3; ++j)
            Rf[i * 3 + j] = R1[0 * 3 + i] * R2[0 * 3 + j] +
                            R1[1 * 3 + i] * R2[1 * 3 + j] +
                            R1[2 * 3 + i] * R2[2 * 3 + j];
    float ui[3];
#pragma unroll
    for (int j = 0; j < 3; ++j)
        ui[j] = -(u1[0] * R1[j * 3 + 0] + u1[1] * R1[j * 3 + 1] + u1[2] * R1[j * 3 + 2]);
    float uf[3];
#pragma unroll
    for (int j = 0; j < 3; ++j)
        uf[j] = ui[0] * R2[0 * 3 + j] + ui[1] * R2[1 * 3 + j] + ui[2] * R2[2 * 3 + j] + u2[j];

    // se3_log(F)
    const float eps = 1e-4f, cos_bound = 1e-4f;
    float tr = Rf[0] + Rf[4] + Rf[8];
    float cphi = fminf(fmaxf(0.5f * (tr - 1.0f), -1.0f + cos_bound), 1.0f - cos_bound);
    float phi = acosf(cphi);
    float s = sinf(phi);
    bool ok = fabsf(s) > 0.5f * eps;
    float fac = ok ? (phi / (2.0f * s)) : (0.5f + phi * phi * (1.0f / 12.0f));
    float w0 = fac * (Rf[2 * 3 + 1] - Rf[1 * 3 + 2]);
    float w1 = fac * (Rf[0 * 3 + 2] - Rf[2 * 3 + 0]);
    float w2 = fac * (Rf[1 * 3 + 0] - Rf[0 * 3 + 1]);
    float w[3] = {w0, w1, w2};
    float ww = w0 * w0 + w1 * w1 + w2 * w2;
    float th2 = fmaxf(ww, eps);
    float th = sqrtf(th2);
    float ca = (1.0f - cosf(th)) / th2;
    float cb = (th - sinf(th)) / (th2 * th);
    // W2 = w w^T - (w.w) I  (exact ww, as in reference)
    float Wm[9] = {0.f, -w2, w1,   w2, 0.f, -w0,   -w1, w0, 0.f};
    float V[9];
#pragma unroll
    for (int i = 0; i < 3; ++i)
#pragma unroll
        for (int j = 0; j < 3; ++j) {
            float id = (i == j) ? 1.0f : 0.0f;
            V[i * 3 + j] = id + ca * Wm[i * 3 + j] + cb * (w[i] * w[j] - ww * id);
        }
    // lt = V^{-1} uf via adjugate / det
    float c00 = V[4] * V[8] - V[5] * V[7];
    float c01 = V[5] * V[6] - V[3] * V[8];
    float c02 = V[3] * V[7] - V[4] * V[6];
    float det = V[0] * c00 + V[1] * c01 + V[2] * c02;
    float rdet = 1.0f / det;
    float inv[9];
    inv[0] = c00 * rdet;
    inv[1] = (V[2] * V[7] - V[1] * V[8]) * rdet;
    inv[2] = (V[1] * V[5] - V[2] * V[4]) * rdet;
    inv[3] = c01 * rdet;
    inv[4] = (V[0] * V[8] - V[2] * V[6]) * rdet;
    inv[5] = (V[2] * V[3] - V[0] * V[5]) * rdet;
    inv[6] = c02 * rdet;
    inv[7] = (V[1] * V[6] - V[0] * V[7]) * rdet;
    inv[8] = (V[0] * V[4] - V[1] * V[3]) * rdet;

    float* o = dof6 + (long long)g * 6;
#pragma unroll
    for (int i = 0; i < 3; ++i)
        o[i] = inv[i * 3 + 0] * uf[0] + inv[i * 3 + 1] * uf[1] + inv[i * 3 + 2] * uf[2];
    o[3] = w0; o[4] = w1; o[5] = w2;
}

// ---------------------------------------------------------------------------
// Streaming WMMA kernel. grid = (chunksPerImg / wavesPerBlock, 2*n).
// Each wave handles 8 tiles of 16 points (128 points).
// A (16x4): A[m][k] = M[k][m] (m=0..2), A[m][3] = u[m]
//   lane m<16:  a0=A[m][0], a1=A[m][1];  lane m+16: a0=A[m][2], a1=A[m][3]
// B (4x16): B[0..2][n] = x,y,z of point n; B[3][n] = 1
//   lane n<16:  b0=x_n, b1=y_n;          lane n+16: b0=z_n, b1=1.0
// Phase 1: issue ALL 8 tile loads (one b64/lane at tp + col*3 + (lane>>4))
//          so they are in flight together (load clause, partial waits).
// Phase 2: per tile, two cndmask selects build B, one WMMA, lanes 0-15
//          store D rows 0..2 as a b96 (contiguous 192B per tile).
// ---------------------------------------------------------------------------
#define TILES_PER_WAVE 8
#define WAVES_PER_BLOCK 8

__global__ __launch_bounds__(256) void tda_transform_kernel(
    const float* __restrict__ X_v, const float* __restrict__ target_X_v,
    const float* __restrict__ mats, float* __restrict__ out, int n, int P) {
    const int lane = threadIdx.x & 31;
    const int col = lane & 15;
    const int hi = lane >> 4;            // 0 for lanes 0-15, 1 for 16-31
    const bool lower = (hi == 0);

    const int img = blockIdx.y;          // [0, 2n)
    const int sel = (img >= n) ? 1 : 0;
    const int g = img - sel * n;
    const int chunk = blockIdx.x * WAVES_PER_BLOCK + (threadIdx.x >> 5);

    // Build A operand (uniform per wave except lane selection; cached loads).
    const float* mb = mats + (long long)img * 12;
    v2f a; a.x = 0.0f; a.y = 0.0f;
    if (col < 3) {
        if (lower) { a.x = mb[0 * 3 + col]; a.y = mb[1 * 3 + col]; }
        else       { a.x = mb[2 * 3 + col]; a.y = mb[3 * 3 + col]; }
    }

    const int ptsPerWave = 16 * TILES_PER_WAVE;  // 128
    const long long base = (long long)g * P * 3 + (long long)chunk * ptsPerWave * 3;
    const float* __restrict__ src = (sel ? target_X_v : X_v) + base;
    float* __restrict__ dst = out + (long long)sel * n * P * 3 + base;

    const float* ldp = src + col * 3 + hi;       // per-lane load base
    float* stp = dst + col * 3;                  // per-lane store base (lanes 0-15)

    // Phase 1: all tile loads in flight (memory-level parallelism in-wave).
    v2f_u p[TILES_PER_WAVE];
#pragma unroll
    for (int t = 0; t < TILES_PER_WAVE; ++t)
        p[t] = __builtin_nontemporal_load((const v2f_u*)(ldp + t * 48));

    // Phase 2: select -> WMMA -> store, draining loads with partial waits.
#pragma unroll
    for (int t = 0; t < TILES_PER_WAVE; ++t) {
        v2f b;
        b.x = lower ? p[t].x : p[t].y;           // x_n | z_n
        b.y = lower ? p[t].y : 1.0f;             // y_n | 1.0
        v8f c = {};
        v8f d = __builtin_amdgcn_wmma_f32_16x16x4_f32(
            /*neg_a=*/false, a, /*neg_b=*/false, b,
            /*c_mod=*/(short)0, c, /*reuse_a=*/false, /*reuse_b=*/false);
        if (lower) {
            float* op = stp + t * 48;
            __builtin_nontemporal_store(d[0], op + 0);
            __builtin_nontemporal_store(d[1], op + 1);
            __builtin_nontemporal_store(d[2], op + 2);
        }
    }
}

extern "C" void kernel_launch(void* const* d_in, const int* in_sizes, int n_in,
                              void* d_out, int out_size, void* d_ws, size_t ws_size,
                              hipStream_t stream) {
    const float* C    = (const float*)d_in[0];
    const float* X_v  = (const float*)d_in[1];
    const float* tX_v = (const float*)d_in[2];
    const float* ang1 = (const float*)d_in[3];
    const float* ang2 = (const float*)d_in[4];
    const float* t1   = (const float*)d_in[5];
    const float* t2   = (const float*)d_in[6];
    float* out = (float*)d_out;

    const int n = in_sizes[0] / 3;                         // 28
    const long long nP = (long long)in_sizes[1] / 3;       // n * P
    const int P = (int)(nP / n);                           // 262144
    float* mats = (float*)d_ws;                            // (2*n)*12 floats

    // dof6 tail lives after the two (n,P,3) outputs
    float* dof6 = out + 2LL * n * P * 3;
    tda_setup_kernel<<<1, 64, 0, stream>>>(C, ang1, ang2, t1, t2, mats, dof6, n);

    const int ptsPerWave = 16 * TILES_PER_WAVE;            // 128
    const int chunksPerImg = P / ptsPerWave;               // 2048
    dim3 grid(chunksPerImg / WAVES_PER_BLOCK, 2 * n);      // (256, 56)
    tda_transform_kernel<<<grid, 256, 0, stream>>>(X_v, tX_v, mats, out, n, P);
}